// Conv2dFusion_88356067213460
// MI455X (gfx1250) — compile-verified
//
#include <hip/hip_runtime.h>

// ---------------------------------------------------------------------------
// Fused Outer2dFusion + 3x ConvBlock for MI455X (gfx1250, wave32, WMMA).
// One workgroup (256 threads = 8 wave32s) per token; all intermediates in LDS.
// Rank-1 factorization of conv0:  out0[c,oy,ox] = U(84x50) * (W0c(50x50) * V(50x84))
// All GEMMs run on V_WMMA_F32_16X16X4_F32 (exact fp32).
// 250 KB LDS => 1 WG/WGP; 8 waves => 2 waves/SIMD32 for latency hiding.
// ---------------------------------------------------------------------------

typedef float v2f __attribute__((ext_vector_type(2)));
typedef float v8f __attribute__((ext_vector_type(8)));

#define NB 8
#define NS 64
#define ND 300
#define NTOK (NB * NS)
#define NTHREADS 256
#define NWAVE 8

// conv0 geometry
#define C0 16
#define K0 50     // 50x50 kernel
#define KP 52     // K padded to multiple of 4
#define OW0 84    // (300-50)/3+1
#define NP 96     // 84 padded to multiple of 16
#define P0 28     // after 3x3/3 pool

// conv1 geometry
#define C1 32
#define OW1 26    // 28-3+1
#define NPIX1 676 // 26*26
#define K1 144    // 16*3*3
#define P1 8

// conv2 geometry
#define C2 32
#define OW2 6     // 8-3+1
#define NPIX2 36
#define NPIX2P 48
#define K2 288    // 32*3*3

// ---- LDS layout (float offsets); regions are reused phase-to-phase --------
#define OFF_VM  0       // [52][96]   im2row of v                (phase 1)
#define OFF_UM  4992    // [96][52]   im2row of u                (phase 1)
#define OFF_W0C 9984    // [64][52]   padded w0 channel          (phase 1)
#define OFF_TC  13312   // [52][96]   t_c = W0c * V              (phase 1)
#define OFF_OC  18304   // [84][96]   conv0 channel output       (phase 1)
#define OFF_X1  26368   // [16][784]  pooled conv0               (phase 1->2)
#define OFF_A1  0       // [32][144]  w1 as GEMM A               (phase 2)
#define OFF_BT  4608    // 8 x [144][16] per-wave im2col tiles -> ends 23040 (phase 2)
#define OFF_O1  38912   // [32][676]  conv1 output               (phase 2)
#define OFF_X2  60544   // [32][64]   pooled conv1               (phase 2->3)
#define OFF_A2  0       // [32][288]  w2 as GEMM A               (phase 3)
#define OFF_B2  9216    // [288][48]  im2col of x2               (phase 3)
#define OFF_O2  23040   // [32][48]   conv2 output               (phase 3)
#define SMEM_FLOATS 62592   // = 250,368 bytes  (< 320 KB/WGP)

__device__ __forceinline__ float gelu_exact(float x) {
  return 0.5f * x * (1.0f + erff(x * 0.70710678118654752f));
}

// A fragment (16x4 fp32): lanes 0-15 -> K=0,1 ; lanes 16-31 -> K=2,3
__device__ __forceinline__ v2f ld_a(const float* A, int lda, int m0, int k0,
                                    int l, int half) {
  return *(const v2f*)(A + (m0 + l) * lda + (k0 + 2 * half));  // lda even, k0 mult of 4
}
// B fragment (4x16 fp32): lanes 0-15 -> K=0,1 ; lanes 16-31 -> K=2,3 ; N = lane&15
__device__ __forceinline__ v2f ld_b(const float* Bm, int ldb, int k0, int n0,
                                    int l, int half) {
  v2f b;
  b.x = Bm[(k0 + 2 * half) * ldb + n0 + l];
  b.y = Bm[(k0 + 2 * half + 1) * ldb + n0 + l];
  return b;
}
// C/D fragment (16x16 fp32): VGPR r -> rows r (lanes 0-15) and r+8 (lanes 16-31)
__device__ __forceinline__ void st_c(float* Cm, int ldc, int m0, int n0, v8f c,
                                     int l, int half, int maxRow, int maxCol) {
  int col = n0 + l;
  if (col >= maxCol) return;
#pragma unroll
  for (int r = 0; r < 8; ++r) {
    int row = m0 + r + 8 * half;
    if (row < maxRow) Cm[row * ldc + col] = c[r];
  }
}

#define WMMA_F32(acc, a, b)                                                   \
  (acc) = __builtin_amdgcn_wmma_f32_16x16x4_f32(false, (a), false, (b),       \
                                                (short)0, (acc), false, false)

extern "C" __global__ __launch_bounds__(NTHREADS) void
fused_outer_conv_kernel(const float* __restrict__ tf,  // token_features [512][300]
                        const float* __restrict__ tp,  // type_preds     [512][300]
                        const float* __restrict__ w0, const float* __restrict__ b0,
                        const float* __restrict__ w1, const float* __restrict__ b1,
                        const float* __restrict__ w2, const float* __restrict__ b2,
                        float* __restrict__ out) {
  extern __shared__ float smem[];
  const int tid  = threadIdx.x;
  const int lane = tid & 31;
  const int l    = tid & 15;
  const int half = (tid >> 4) & 1;
  const int wave = tid >> 5;
  const int tok  = blockIdx.x;

  const float* v = tf + tok * ND;  // columns (j index of outer product)
  const float* u = tp + tok * ND;  // rows    (i index of outer product)

  // Warm the later-phase weights into cache while phase 1 runs
  // (global_prefetch_b8; one 128 B line per thread covers all of w1, only the first 32 KB of w2).
  {
    const char* pw1 = (const char*)w1;
    const char* pw2 = (const char*)w2;
    if (tid * 128 < (int)(C1 * K1 * sizeof(float))) __builtin_prefetch(pw1 + tid * 128, 0, 1);
    if (tid * 128 < (int)(C2 * K2 * sizeof(float))) __builtin_prefetch(pw2 + tid * 128, 0, 1);
  }

  float* Vm  = smem + OFF_VM;
  float* Um  = smem + OFF_UM;
  float* W0c = smem + OFF_W0C;
  float* Tc  = smem + OFF_TC;
  float* Oc  = smem + OFF_OC;
  float* X1  = smem + OFF_X1;

  // ---- Phase 1: build padded im2row matrices for the rank-1 conv0 --------
  for (int i = tid; i < KP * NP; i += NTHREADS) {
    int kx = i / NP, ox = i % NP;
    Vm[i] = (kx < K0 && ox < OW0) ? v[3 * ox + kx] : 0.f;
  }
  for (int i = tid; i < NP * KP; i += NTHREADS) {
    int oy = i / KP, ky = i % KP;
    Um[i] = (oy < OW0 && ky < K0) ? u[3 * oy + ky] : 0.f;
  }

  for (int c = 0; c < C0; ++c) {
    __syncthreads();
    // padded channel weights [64][52]
    for (int i = tid; i < 64 * KP; i += NTHREADS) {
      int ky = i / KP, kx = i % KP;
      W0c[i] = (ky < K0 && kx < K0) ? w0[(c * K0 + ky) * K0 + kx] : 0.f;
    }
    __syncthreads();

    // Stage A: Tc(50x84) = W0c(50x50) * Vm(50x84)   [padded 64x52 * 52x96]
    for (int t = wave; t < 4 * 6; t += NWAVE) {
      int m0 = (t / 6) * 16, n0 = (t % 6) * 16;
      v8f acc = {0.f, 0.f, 0.f, 0.f, 0.f, 0.f, 0.f, 0.f};
      for (int k0 = 0; k0 < KP; k0 += 4) {
        v2f a  = ld_a(W0c, KP, m0, k0, l, half);
        v2f bb = ld_b(Vm, NP, k0, n0, l, half);
        WMMA_F32(acc, a, bb);
      }
      st_c(Tc, NP, m0, n0, acc, l, half, KP, NP);  // rows 50,51 land as zeros
    }
    __syncthreads();

    // Stage B: Oc(84x84) = Um(84x50) * Tc(50x84)   [padded 96x52 * 52x96]
    for (int t = wave; t < 6 * 6; t += NWAVE) {
      int m0 = (t / 6) * 16, n0 = (t % 6) * 16;
      v8f acc = {0.f, 0.f, 0.f, 0.f, 0.f, 0.f, 0.f, 0.f};
      for (int k0 = 0; k0 < KP; k0 += 4) {
        v2f a  = ld_a(Um, KP, m0, k0, l, half);
        v2f bb = ld_b(Tc, NP, k0, n0, l, half);
        WMMA_F32(acc, a, bb);
      }
      st_c(Oc, NP, m0, n0, acc, l, half, OW0, NP);
    }
    __syncthreads();

    // bias + exact GELU + 3x3/3 maxpool -> X1[c]
    float bc = b0[c];
    for (int i = tid; i < P0 * P0; i += NTHREADS) {
      int py = i / P0, px = i % P0;
      float m = -3.402823466e38f;
      for (int dy = 0; dy < 3; ++dy)
        for (int dx = 0; dx < 3; ++dx)
          m = fmaxf(m, gelu_exact(Oc[(3 * py + dy) * NP + (3 * px + dx)] + bc));
      X1[c * (P0 * P0) + i] = m;
    }
  }
  __syncthreads();

  // ---- Phase 2: conv1 as implicit GEMM  (32 x 676, K=144) -----------------
  float* A1 = smem + OFF_A1;
  float* O1 = smem + OFF_O1;
  float* X2 = smem + OFF_X2;
  for (int i = tid; i < C1 * K1; i += NTHREADS) A1[i] = w1[i];  // [co][ci*9+ky*3+kx]
  __syncthreads();

  float* Bt = smem + OFF_BT + wave * (K1 * 16);  // private per-wave im2col slab
  for (int nt = wave; nt < (NPIX1 + 15) / 16; nt += NWAVE) {
    for (int i = lane; i < K1 * 16; i += 32) {
      int kk = i >> 4, j = i & 15;
      int p = nt * 16 + j;
      float val = 0.f;
      if (p < NPIX1) {
        int py = p / OW1, px = p % OW1;
        int ci = kk / 9, r9 = kk % 9;
        val = X1[ci * (P0 * P0) + (py + r9 / 3) * P0 + (px + r9 % 3)];
      }
      Bt[i] = val;  // per-wave LDS is in-order; visible to this wave's WMMA loads
    }
    for (int m0 = 0; m0 < C1; m0 += 16) {
      v8f acc = {0.f, 0.f, 0.f, 0.f, 0.f, 0.f, 0.f, 0.f};
      for (int k0 = 0; k0 < K1; k0 += 4) {
        v2f a  = ld_a(A1, K1, m0, k0, l, half);
        v2f bb = ld_b(Bt, 16, k0, 0, l, half);
        WMMA_F32(acc, a, bb);
      }
      int col = nt * 16 + l;
      if (col < NPIX1) {
#pragma unroll
        for (int r = 0; r < 8; ++r) O1[(m0 + r + 8 * half) * NPIX1 + col] = acc[r];
      }
    }
  }
  __syncthreads();

  // bias + GELU + 3x3/3 pool -> X2[32][8][8]
  for (int i = tid; i < C1 * P1 * P1; i += NTHREADS) {
    int co = i >> 6, rem = i & 63, py = rem >> 3, px = rem & 7;
    float bc = b1[co];
    float m = -3.402823466e38f;
    for (int dy = 0; dy < 3; ++dy)
      for (int dx = 0; dx < 3; ++dx)
        m = fmaxf(m, gelu_exact(O1[co * NPIX1 + (3 * py + dy) * OW1 + (3 * px + dx)] + bc));
    X2[i] = m;
  }
  __syncthreads();

  // ---- Phase 3: conv2 as GEMM  (32 x 36->48, K=288) -----------------------
  float* A2 = smem + OFF_A2;
  float* B2 = smem + OFF_B2;
  float* O2 = smem + OFF_O2;
  for (int i = tid; i < C2 * K2; i += NTHREADS) A2[i] = w2[i];
  for (int i = tid; i < K2 * NPIX2P; i += NTHREADS) {
    int kk = i / NPIX2P, col = i % NPIX2P;
    float val = 0.f;
    if (col < NPIX2) {
      int py = col / OW2, px = col % OW2;
      int ci = kk / 9, r9 = kk % 9;
      val = X2[ci * (P1 * P1) + (py + r9 / 3) * P1 + (px + r9 % 3)];
    }
    B2[i] = val;
  }
  __syncthreads();

  for (int t = wave; t < 2 * 3; t += NWAVE) {
    int m0 = (t / 3) * 16, n0 = (t % 3) * 16;
    v8f acc = {0.f, 0.f, 0.f, 0.f, 0.f, 0.f, 0.f, 0.f};
    for (int k0 = 0; k0 < K2; k0 += 4) {
      v2f a  = ld_a(A2, K2, m0, k0, l, half);
      v2f bb = ld_b(B2, NPIX2P, k0, n0, l, half);
      WMMA_F32(acc, a, bb);
    }
    st_c(O2, NPIX2P, m0, n0, acc, l, half, C2, NPIX2P);
  }
  __syncthreads();

  // bias + GELU + 3x3/3 pool over 6x6 -> 2x2 ; write 128 outputs per token
  if (tid < 128) {
    int co = tid >> 2, rem = tid & 3, py = rem >> 1, px = rem & 1;
    float bc = b2[co];
    float m = -3.402823466e38f;
    for (int dy = 0; dy < 3; ++dy)
      for (int dx = 0; dx < 3; ++dx)
        m = fmaxf(m, gelu_exact(O2[co * NPIX2P + (3 * py + dy) * OW2 + (3 * px + dx)] + bc));
    out[tok * 128 + tid] = m;
  }
}

extern "C" void kernel_launch(void* const* d_in, const int* in_sizes, int n_in,
                              void* d_out, int out_size, void* d_ws, size_t ws_size,
                              hipStream_t stream) {
  (void)in_sizes; (void)n_in; (void)out_size; (void)d_ws; (void)ws_size;
  const float* tf = (const float*)d_in[0];  // token_features [8][64][300]
  const float* tp = (const float*)d_in[1];  // type_preds     [8][64][300]
  const float* w0 = (const float*)d_in[2];
  const float* b0 = (const float*)d_in[3];
  const float* w1 = (const float*)d_in[4];
  const float* b1 = (const float*)d_in[5];
  const float* w2 = (const float*)d_in[6];
  const float* b2 = (const float*)d_in[7];
  float* out = (float*)d_out;

  const size_t shmem = SMEM_FLOATS * sizeof(float);  // ~250 KB of 320 KB/WGP
  (void)hipFuncSetAttribute((const void*)fused_outer_conv_kernel,
                            hipFuncAttributeMaxDynamicSharedMemorySize, (int)shmem);
  hipLaunchKernelGGL(fused_outer_conv_kernel, dim3(NTOK), dim3(NTHREADS), shmem,
                     stream, tf, tp, w0, b0, w1, b1, w2, b2, out);
}